// DynamicSphericalTorch_3032246911173
// MI455X (gfx1250) — compile-verified
//
#include <hip/hip_runtime.h>
#include <cstdint>
#include <cstddef>

// Problem constants (match reference).
constexpr int kB    = 64;        // batch
constexpr int kN    = 200000;    // nodes
constexpr int kS    = 6;         // steps
constexpr int kE    = 1000000;   // edges per step
constexpr int kNIn  = 4096;
constexpr int kNOut = 1024;

constexpr int kThreads       = 256;  // 8 waves (wave32)
constexpr int kEdgesPerBlock = 512;  // 64 edges per wave
constexpr int kWavesPerBlock = kThreads / 32;
constexpr int kEdgesPerWave  = kEdgesPerBlock / kWavesPerBlock;
constexpr int kTotal4        = kN * kB / 4;   // 3,200,000 float4 groups

// ---------------------------------------------------------------------------
// init: state[n,b] = tanh(x[b,n]*w_in[n]+bias[n]) for n<N_IN else 0;
// also zero msg and fan-in flags. float4-vectorized (16 threads per node row).
// ---------------------------------------------------------------------------
__global__ __launch_bounds__(256) void init_state(
    const float* __restrict__ x, const float* __restrict__ w_in,
    const float* __restrict__ bias, float* __restrict__ state,
    float* __restrict__ msg, int* __restrict__ flag) {
  const int i = blockIdx.x * blockDim.x + threadIdx.x;
  if (i >= kTotal4) return;
  const int    n = i >> 4;            // B/4 == 16 groups per node
  const size_t j = (size_t)i * 4;
  float4 v = make_float4(0.f, 0.f, 0.f, 0.f);
  if (n < kNIn) {
    const int   b0 = (int)(j & 63);
    const float wn = w_in[n];
    const float bn = bias[n];
    v.x = tanhf(x[(size_t)(b0 + 0) * kNIn + n] * wn + bn);
    v.y = tanhf(x[(size_t)(b0 + 1) * kNIn + n] * wn + bn);
    v.z = tanhf(x[(size_t)(b0 + 2) * kNIn + n] * wn + bn);
    v.w = tanhf(x[(size_t)(b0 + 3) * kNIn + n] * wn + bn);
  }
  *(float4*)(state + j) = v;
  *(float4*)(msg + j)   = make_float4(0.f, 0.f, 0.f, 0.f);
  if (i < kN / 4) ((int4*)flag)[i] = make_int4(0, 0, 0, 0);
}

// ---------------------------------------------------------------------------
// Edge scatter: one wave per edge; lane l owns batch elems {2l, 2l+1}.
// Edge metadata staged to LDS via CDNA5 async global->LDS B128 loads
// (ASYNCcnt), then each wave streams its 64 edges:
//   gather state[src,:] (256B row, L2-resident), scale, hardware
//   global_atomic_add_f32 into msg[dst,:], benign-race fan-in flag store.
// ---------------------------------------------------------------------------
__global__ __launch_bounds__(kThreads) void edge_scatter(
    const float* __restrict__ w_edge, const int* __restrict__ src,
    const int* __restrict__ dst, const float* __restrict__ state,
    float* __restrict__ msg, int* __restrict__ flag) {
  __shared__ __align__(16) float lw[kEdgesPerBlock];
  __shared__ __align__(16) int   ls[kEdgesPerBlock];
  __shared__ __align__(16) int   ld[kEdgesPerBlock];

  const int  tid  = threadIdx.x;
  const long base = (long)blockIdx.x * kEdgesPerBlock;

  // --- async stage edge tables into LDS, 16B per lane (ASYNCcnt path) ---
  // 3 arrays x 128 four-dword chunks. E % 4 == 0 and chunk starts are
  // 4-aligned, so a chunk is either fully valid or fully out of range.
  for (int t = tid; t < 3 * (kEdgesPerBlock / 4); t += kThreads) {
    const int  arr = t >> 7;           // 0: w_edge, 1: src, 2: dst
    const int  c   = (t & 127) << 2;   // chunk start within block
    const long e   = base + c;
    if (e < kE) {
      uint32_t    a;
      const void* g;
      if (arr == 0)      { a = (uint32_t)(uintptr_t)&lw[c]; g = w_edge + e; }
      else if (arr == 1) { a = (uint32_t)(uintptr_t)&ls[c]; g = src + e; }
      else               { a = (uint32_t)(uintptr_t)&ld[c]; g = dst + e; }
      asm volatile("global_load_async_to_lds_b128 %0, %1, off"
                   :: "v"(a), "v"(g) : "memory");
    }
  }
  asm volatile("s_wait_asynccnt 0" ::: "memory");
  __syncthreads();

  const int lane   = tid & 31;            // wave32
  const int waveId = tid >> 5;
  const int b0     = lane * 2;            // 32 lanes x float2 == B=64 row
  const int eStart = waveId * kEdgesPerWave;

  // number of valid edges for this wave
  int cnt = (int)(kE - base) - eStart;
  if (cnt > kEdgesPerWave) cnt = kEdgesPerWave;
  if (cnt < 0) cnt = 0;

  auto body = [&](int k) {
    const int eL = eStart + k;
    // lookahead prefetch of a future gather row toward L0
    if (k + 8 < cnt) {
      const int sPF = ls[eL + 8];
      __builtin_prefetch(&state[(size_t)sPF * kB + b0], 0, 1);
    }
    // uniform per-edge metadata -> SGPRs (saddr-form gathers)
    const int   s = __builtin_amdgcn_readfirstlane(ls[eL]);
    const int   d = __builtin_amdgcn_readfirstlane(ld[eL]);
    const float w = __uint_as_float(
        __builtin_amdgcn_readfirstlane(__float_as_uint(lw[eL])));

    const float2 v =
        *reinterpret_cast<const float2*>(&state[(size_t)s * kB + b0]);
    float* mp = &msg[(size_t)d * kB + b0];
    unsafeAtomicAdd(mp,     w * v.x);     // global_atomic_add_f32 (no return)
    unsafeAtomicAdd(mp + 1, w * v.y);
    if (lane == 0) flag[d] = 1;           // benign race: all writers store 1
  };

  if (cnt == kEdgesPerWave) {             // full block (all but the tail)
#pragma unroll 4
    for (int k = 0; k < kEdgesPerWave; ++k) body(k);
  } else {
    for (int k = 0; k < cnt; ++k) body(k);
  }
}

// ---------------------------------------------------------------------------
// update: nodes with incoming edges this step; re-zeroes the msg lanes it
// consumed (untouched nodes still hold 0), so no full clear pass is needed.
// ---------------------------------------------------------------------------
__global__ __launch_bounds__(256) void update_state(
    const float* __restrict__ bias, float* __restrict__ msg,
    const int* __restrict__ flag, float* __restrict__ state) {
  const int i = blockIdx.x * blockDim.x + threadIdx.x;
  if (i >= kTotal4) return;
  const int n = i >> 4;
  if (!flag[n]) return;
  const size_t j = (size_t)i * 4;
  float4 m = *(float4*)(msg + j);
  float4 s = *(float4*)(state + j);
  const float bn = bias[n];
  s.x = tanhf(m.x + bn + s.x);
  s.y = tanhf(m.y + bn + s.y);
  s.z = tanhf(m.z + bn + s.z);
  s.w = tanhf(m.w + bn + s.w);
  *(float4*)(state + j) = s;
  *(float4*)(msg + j)   = make_float4(0.f, 0.f, 0.f, 0.f);
}

// ---------------------------------------------------------------------------
// clear fan-in flags between steps (0.8 MB instead of a 51 MB msg sweep)
// ---------------------------------------------------------------------------
__global__ __launch_bounds__(256) void flag_clear(int* __restrict__ flag) {
  const int i = blockIdx.x * blockDim.x + threadIdx.x;
  if (i < kN / 4) ((int4*)flag)[i] = make_int4(0, 0, 0, 0);
}

// ---------------------------------------------------------------------------
// output: out[b, j] = state[N - N_OUT + j, b]
// ---------------------------------------------------------------------------
__global__ __launch_bounds__(256) void gather_out(
    const float* __restrict__ state, float* __restrict__ out) {
  const int i = blockIdx.x * blockDim.x + threadIdx.x;
  if (i >= kB * kNOut) return;
  const int b = i >> 10;       // N_OUT == 1024
  const int j = i & 1023;
  out[i] = state[(size_t)(kN - kNOut + j) * kB + b];
}

// ---------------------------------------------------------------------------
extern "C" void kernel_launch(void* const* d_in, const int* in_sizes, int n_in,
                              void* d_out, int out_size, void* d_ws,
                              size_t ws_size, hipStream_t stream) {
  (void)in_sizes; (void)n_in; (void)out_size; (void)ws_size;

  const float* x      = (const float*)d_in[0];  // [B, N_IN]
  const float* w_in   = (const float*)d_in[1];  // [N_IN]
  const float* bias   = (const float*)d_in[2];  // [N]
  const float* w_edge = (const float*)d_in[3];  // [S, E]
  const int*   src    = (const int*)d_in[4];    // [S, E]
  const int*   dst    = (const int*)d_in[5];    // [S, E]
  float*       out    = (float*)d_out;          // [B, N_OUT]

  // workspace layout: state | msg | flag  (~103 MB)
  float* state = (float*)d_ws;
  float* msg   = state + (size_t)kN * kB;
  int*   flag  = (int*)(msg + (size_t)kN * kB);

  const unsigned gz4 = (unsigned)((kTotal4 + 255) / 256);          // 12500
  const int edgeBlocks = (kE + kEdgesPerBlock - 1) / kEdgesPerBlock; // 1954

  init_state<<<gz4, 256, 0, stream>>>(x, w_in, bias, state, msg, flag);

  for (int step = 0; step < kS; ++step) {
    edge_scatter<<<edgeBlocks, kThreads, 0, stream>>>(
        w_edge + (size_t)step * kE, src + (size_t)step * kE,
        dst + (size_t)step * kE, state, msg, flag);
    update_state<<<gz4, 256, 0, stream>>>(bias, msg, flag, state);
    if (step + 1 < kS)
      flag_clear<<<(kN / 4 + 255) / 256, 256, 0, stream>>>(flag);
  }

  gather_out<<<(kB * kNOut + 255) / 256, 256, 0, stream>>>(state, out);
}